// DeformableBEVRefine_30356828848263
// MI455X (gfx1250) — compile-verified
//
#include <hip/hip_runtime.h>
#include <math.h>

typedef float v2f __attribute__((ext_vector_type(2)));
typedef float v8f __attribute__((ext_vector_type(8)));

#define BB 2
#define DD 128
#define HH 256
#define WW 256
#define HWT (HH*WW)          // 65536
#define PP  (BB*HWT)         // 131072
#define NPART 1024

// ---------------------------------------------------------------------------
// Kernel 1: fused 3x3 convs (offsets: 8ch, attn logits: 4ch) + softmax over K
// 16x16 pixel tile per block, halo'd 18x18 patch in LDS, 32-channel chunks.
// ---------------------------------------------------------------------------
__global__ __launch_bounds__(256) void conv_off_wgt(
    const float* __restrict__ bev,
    const float* __restrict__ offw, const float* __restrict__ offb,
    const float* __restrict__ wgtw, const float* __restrict__ wgtb,
    float* __restrict__ offs, float* __restrict__ attn)
{
    __shared__ float patch[32 * 18 * 18];   // [ci][y][x]   41472 B
    __shared__ float wsm[12 * 32 * 9];      // [co][ci][k]  13824 B
    const int tid = threadIdx.x;
    const int tx = tid & 15, ty = tid >> 4;
    const int gx = blockIdx.x * 16 + tx;
    const int gy = blockIdx.y * 16 + ty;
    const int b  = blockIdx.z;

    float acc[12];
#pragma unroll
    for (int i = 0; i < 12; ++i) acc[i] = 0.f;

    for (int cc = 0; cc < DD; cc += 32) {
        // stage halo'd input patch (zero padding outside image: SAME conv)
        for (int idx = tid; idx < 32 * 18 * 18; idx += 256) {
            int ci = idx / (18 * 18);
            int r  = idx - ci * (18 * 18);
            int py = r / 18, px = r - py * 18;
            int iy = blockIdx.y * 16 + py - 1;
            int ix = blockIdx.x * 16 + px - 1;
            float v = 0.f;
            if (iy >= 0 && iy < HH && ix >= 0 && ix < WW)
                v = bev[(((size_t)b * DD + cc + ci) * HH + iy) * WW + ix];
            patch[idx] = v;
        }
        // stage weights for this channel chunk (offset_w: 8 ch, weight_w: 4 ch)
        for (int idx = tid; idx < 12 * 32 * 9; idx += 256) {
            int co = idx / (32 * 9);
            int r  = idx - co * (32 * 9);
            int ci = r / 9, k = r - ci * 9;
            float v;
            if (co < 8) v = offw[(co * DD + cc + ci) * 9 + k];
            else        v = wgtw[((co - 8) * DD + cc + ci) * 9 + k];
            wsm[idx] = v;
        }
        __syncthreads();

        for (int ci = 0; ci < 32; ++ci) {
            float p[9];
#pragma unroll
            for (int dy = 0; dy < 3; ++dy)
#pragma unroll
                for (int dx = 0; dx < 3; ++dx)
                    p[dy * 3 + dx] = patch[ci * (18 * 18) + (ty + dy) * 18 + (tx + dx)];
            const float* wrow = &wsm[ci * 9];
#pragma unroll
            for (int co = 0; co < 12; ++co) {
                float s = 0.f;
#pragma unroll
                for (int k = 0; k < 9; ++k) s += p[k] * wrow[co * 32 * 9 + k];
                acc[co] += s;
            }
        }
        __syncthreads();
    }

    const size_t p = (size_t)gy * WW + gx;
#pragma unroll
    for (int co = 0; co < 8; ++co)
        offs[((size_t)b * 8 + co) * HWT + p] = acc[co] + offb[co];

    float l[4], m = -1e30f;
#pragma unroll
    for (int k = 0; k < 4; ++k) { l[k] = acc[8 + k] + wgtb[k]; m = fmaxf(m, l[k]); }
    float s = 0.f;
#pragma unroll
    for (int k = 0; k < 4; ++k) { l[k] = __expf(l[k] - m); s += l[k]; }
    float inv = 1.f / s;
#pragma unroll
    for (int k = 0; k < 4; ++k)
        attn[((size_t)b * 4 + k) * HWT + p] = l[k] * inv;
}

// ---------------------------------------------------------------------------
// Kernel 2: bilinear sample (border clamp) * attn, summed over K.
// One wave per pixel; each lane handles channels {lane, lane+32, +64, +96}.
// NOTE the reference builds its base grid transposed: sample x-coord = row+dx,
// sample y-coord = col+dy (H==W so shapes line up). Output stored NHWC so it
// is directly the GEMM A matrix. bev (67 MB) is L2-resident (192 MB L2), so
// the random gathers hit L2, not HBM.
// ---------------------------------------------------------------------------
__global__ __launch_bounds__(256) void sample_agg(
    const float* __restrict__ bev,
    const float* __restrict__ offs, const float* __restrict__ attn,
    float* __restrict__ agg)
{
    const int tid  = threadIdx.x;
    const int lane = tid & 31;
    const int wv   = tid >> 5;
    const size_t pix = (size_t)blockIdx.x * 8 + wv;
    const int b  = (int)(pix / HWT);
    const int hw = (int)(pix % HWT);
    const int y  = hw / WW;
    const int x  = hw - y * WW;

    const float* bevb = bev + (size_t)b * DD * HWT;
    float a0 = 0.f, a1 = 0.f, a2 = 0.f, a3 = 0.f;

#pragma unroll
    for (int k = 0; k < 4; ++k) {
        float dx = offs[((size_t)b * 8 + 2 * k + 0) * HWT + hw];
        float dy = offs[((size_t)b * 8 + 2 * k + 1) * HWT + hw];
        float a  = attn[((size_t)b * 4 + k) * HWT + hw];
        // reference: px = clamp(row + dx), py = clamp(col + dy)
        float px = fminf(fmaxf((float)y + dx, 0.f), (float)(WW - 1));
        float py = fminf(fmaxf((float)x + dy, 0.f), (float)(HH - 1));
        float x0f = floorf(px), y0f = floorf(py);
        float wx = px - x0f,  wy = py - y0f;
        int x0 = (int)x0f, y0 = (int)y0f;
        int x1 = min(x0 + 1, WW - 1), y1 = min(y0 + 1, HH - 1);
        float w00 = a * (1.f - wx) * (1.f - wy);
        float w01 = a * wx * (1.f - wy);
        float w10 = a * (1.f - wx) * wy;
        float w11 = a * wx * wy;
        size_t i00 = (size_t)y0 * WW + x0;
        size_t i01 = (size_t)y0 * WW + x1;
        size_t i10 = (size_t)y1 * WW + x0;
        size_t i11 = (size_t)y1 * WW + x1;
        {
            const float* pl = bevb + (size_t)(lane +  0) * HWT;
            a0 += w00 * pl[i00] + w01 * pl[i01] + w10 * pl[i10] + w11 * pl[i11];
        }
        {
            const float* pl = bevb + (size_t)(lane + 32) * HWT;
            a1 += w00 * pl[i00] + w01 * pl[i01] + w10 * pl[i10] + w11 * pl[i11];
        }
        {
            const float* pl = bevb + (size_t)(lane + 64) * HWT;
            a2 += w00 * pl[i00] + w01 * pl[i01] + w10 * pl[i10] + w11 * pl[i11];
        }
        {
            const float* pl = bevb + (size_t)(lane + 96) * HWT;
            a3 += w00 * pl[i00] + w01 * pl[i01] + w10 * pl[i10] + w11 * pl[i11];
        }
    }
    float* dst = agg + pix * DD;
    dst[lane +  0] = a0;
    dst[lane + 32] = a1;
    dst[lane + 64] = a2;
    dst[lane + 96] = a3;
}

// ---------------------------------------------------------------------------
// Kernel 3: proj 1x1 conv as fp32 WMMA GEMM.
// y[p][co] = sum_ci agg[p][ci] * proj_w[co][ci]
// Block = 8 waves x 64 pixels. Wave w owns output columns [16w,16w+16) of
// FOUR 16-pixel M-tiles: the B fragment (proj_w) is loaded once per k-step
// and reused by 4 independent WMMA accumulator chains -> breaks the
// WMMA->WMMA RAW hazard chain and cuts the load:WMMA ratio to 1.25.
// K=128 swept in steps of 4 -> 128 x v_wmma_f32_16x16x4_f32 per wave.
// Operand layouts per CDNA5 ISA 7.12.2:
//   A 16x4 : lane L holds A[M=L%16][K = khalf + {0,1}], khalf = 2*(L>=16)
//   B 4x16 : lane L holds B[K = khalf + {0,1}][N = L%16] = proj_w[N][K] pair
//   D 16x16: vgpr r, lane L -> (M = r + 8*(L>=16), N = L%16)
// ---------------------------------------------------------------------------
__global__ __launch_bounds__(256) void proj_gemm(
    const float* __restrict__ agg,
    const float* __restrict__ projw,
    float* __restrict__ yout)
{
    const int tid   = threadIdx.x;
    const int lane  = tid & 31;
    const int wv    = tid >> 5;               // 0..7 -> n0 = 16*wv
    const int mrow  = lane & 15;
    const int khalf = (lane >> 4) << 1;       // 0 or 2
    const size_t pixBase = (size_t)blockIdx.x * 64;

    const float* brow = projw + (size_t)(wv * 16 + mrow) * DD + khalf;
    const float* a0p  = agg + (pixBase + mrow) * DD + khalf;
    const float* a1p  = a0p + (size_t)16 * DD;
    const float* a2p  = a0p + (size_t)32 * DD;
    const float* a3p  = a0p + (size_t)48 * DD;

    // CDNA5 prefetch (global_prefetch_b8): warm next block's A region in L2.
    __builtin_prefetch(a0p + (size_t)64 * DD, 0, 0);
    __builtin_prefetch(a1p + (size_t)64 * DD, 0, 0);
    __builtin_prefetch(a2p + (size_t)64 * DD, 0, 0);
    __builtin_prefetch(a3p + (size_t)64 * DD, 0, 0);

    v8f c0 = {}, c1 = {}, c2 = {}, c3 = {};
#pragma unroll
    for (int kk = 0; kk < DD; kk += 4) {
        v2f bv = *(const v2f*)(brow + kk);
        v2f a0 = *(const v2f*)(a0p + kk);
        v2f a1 = *(const v2f*)(a1p + kk);
        v2f a2 = *(const v2f*)(a2p + kk);
        v2f a3 = *(const v2f*)(a3p + kk);
        c0 = __builtin_amdgcn_wmma_f32_16x16x4_f32(false, a0, false, bv, (short)0, c0, false, false);
        c1 = __builtin_amdgcn_wmma_f32_16x16x4_f32(false, a1, false, bv, (short)0, c1, false, false);
        c2 = __builtin_amdgcn_wmma_f32_16x16x4_f32(false, a2, false, bv, (short)0, c2, false, false);
        c3 = __builtin_amdgcn_wmma_f32_16x16x4_f32(false, a3, false, bv, (short)0, c3, false, false);
    }

    const int nn    = wv * 16 + (lane & 15);
    const int mbase = (lane >> 4) << 3;       // 0 or 8
    float* d0 = yout + (pixBase +  0 + mbase) * DD + nn;
    float* d1 = yout + (pixBase + 16 + mbase) * DD + nn;
    float* d2 = yout + (pixBase + 32 + mbase) * DD + nn;
    float* d3 = yout + (pixBase + 48 + mbase) * DD + nn;
#pragma unroll
    for (int r = 0; r < 8; ++r) {
        d0[(size_t)r * DD] = c0[r];
        d1[(size_t)r * DD] = c1[r];
        d2[(size_t)r * DD] = c2[r];
        d3[(size_t)r * DD] = c3[r];
    }
}

// ---------------------------------------------------------------------------
// Kernels 4/5: deterministic per-channel mean/var reduction over (B,H,W).
// ---------------------------------------------------------------------------
__global__ __launch_bounds__(256) void bn_partial(
    const float* __restrict__ y, float* __restrict__ part)
{
    const int tid = threadIdx.x;
    const int c   = tid & 127;
    const int sub = tid >> 7;                 // 0 or 1
    float s = 0.f, s2 = 0.f;
    for (size_t p = (size_t)blockIdx.x * 2 + sub; p < PP; p += 2 * NPART) {
        float v = y[p * DD + c];
        s += v; s2 += v * v;
    }
    __shared__ float ls[256], ls2[256];
    ls[tid] = s; ls2[tid] = s2;
    __syncthreads();
    if (sub == 0) {
        part[(size_t)blockIdx.x * DD + c]           = ls[c] + ls[c + 128];
        part[(size_t)(NPART + blockIdx.x) * DD + c] = ls2[c] + ls2[c + 128];
    }
}

__global__ void bn_reduce(const float* __restrict__ part, float* __restrict__ stats)
{
    const int c = threadIdx.x;                // 128 threads
    float s = 0.f, s2 = 0.f;
    for (int i = 0; i < NPART; ++i) {
        s  += part[(size_t)i * DD + c];
        s2 += part[(size_t)(NPART + i) * DD + c];
    }
    float mean = s * (1.f / (float)PP);
    float var  = s2 * (1.f / (float)PP) - mean * mean;
    stats[c]       = mean;
    stats[128 + c] = rsqrtf(var + 1e-5f);
}

// ---------------------------------------------------------------------------
// Kernel 6: BN affine + exact GELU, NHWC -> NCHW transpose into d_out.
// Consecutive threads -> consecutive pixels, fixed channel: coalesced writes.
// ---------------------------------------------------------------------------
__global__ __launch_bounds__(256) void bn_gelu_out(
    const float* __restrict__ y, const float* __restrict__ stats,
    const float* __restrict__ gamma, const float* __restrict__ beta,
    float* __restrict__ out)
{
    const size_t gid = (size_t)blockIdx.x * 256 + threadIdx.x;  // over PP*DD
    const size_t p = gid % PP;
    const int    c = (int)(gid / PP);
    float v = y[p * DD + c];
    v = (v - stats[c]) * stats[128 + c];
    v = v * gamma[c] + beta[c];
    float g = 0.5f * v * (1.f + erff(v * 0.70710678118654752f));
    const size_t b = p / HWT, hw = p % HWT;
    out[((size_t)b * DD + c) * HWT + hw] = g;
}

// ---------------------------------------------------------------------------
extern "C" void kernel_launch(void* const* d_in, const int* in_sizes, int n_in,
                              void* d_out, int out_size, void* d_ws, size_t ws_size,
                              hipStream_t stream) {
    (void)in_sizes; (void)n_in; (void)out_size; (void)ws_size;
    const float* bev   = (const float*)d_in[0];
    const float* offw  = (const float*)d_in[1];
    const float* offb  = (const float*)d_in[2];
    const float* wgtw  = (const float*)d_in[3];
    const float* wgtb  = (const float*)d_in[4];
    const float* projw = (const float*)d_in[5];
    const float* gamma = (const float*)d_in[6];
    const float* beta  = (const float*)d_in[7];

    float* ws    = (float*)d_ws;
    float* offs  = ws;                                  //  4 MB
    float* attn  = offs + (size_t)BB * 8 * HWT;         //  2 MB
    float* agg   = attn + (size_t)BB * 4 * HWT;         // 64 MB (NHWC)
    float* yb    = agg  + (size_t)PP * DD;              // 64 MB (NHWC)
    float* part  = yb   + (size_t)PP * DD;              //  1 MB
    float* stats = part + (size_t)2 * NPART * DD;       //  1 KB
    float* out   = (float*)d_out;

    dim3 g1(WW / 16, HH / 16, BB);
    conv_off_wgt<<<g1, 256, 0, stream>>>(bev, offw, offb, wgtw, wgtb, offs, attn);
    sample_agg  <<<PP / 8, 256, 0, stream>>>(bev, offs, attn, agg);
    proj_gemm   <<<PP / 64, 256, 0, stream>>>(agg, projw, yb);
    bn_partial  <<<NPART, 256, 0, stream>>>(yb, part);
    bn_reduce   <<<1, 128, 0, stream>>>(part, stats);
    bn_gelu_out <<<(PP * (size_t)DD) / 256, 256, 0, stream>>>(yb, stats, gamma, beta, out);
}